// KDLayerNorm_27487790694589
// MI455X (gfx1250) — compile-verified
//
#include <hip/hip_runtime.h>
#include <math.h>

// KD "quantile normalization" layer:
//   per row (D=256): bw = 0.9*std(ddof=1)*D^-0.2
//   cdf_i = mean_j 0.5*(1+erf((x_i-x_j)/(bw*sqrt2)))
//   out_i = erfinv(2*cdf_i-1)*sqrt2 * weight_i + bias_i
//
// Compute-bound: N*D^2 = 268M erf evals vs only 8 MB of HBM traffic
// (0.35us at 23.3 TB/s) -> pure transcendental-VALU roofline. WMMA has no
// honest role: erf is elementwise-nonlinear on pairwise differences, not
// factorable into GEMM. CDNA5 paths used: async global->LDS row staging
// (ASYNCcnt) + LDS broadcast reuse; branchless erf rides the TRANS pipe.

#define KD_D 256

// Branchless erf, abs error ~1.5e-7 (Abramowitz & Stegun 7.1.26).
// erf(x) = sign(x) * (1 - (a1 t + ... + a5 t^5) e^{-x^2}),  t = 1/(1+p|x|)
__device__ __forceinline__ float fast_erf(float x) {
    const float ax = fabsf(x);
    const float t  = __builtin_amdgcn_rcpf(fmaf(0.3275911f, ax, 1.0f)); // v_rcp_f32
    float p = fmaf(1.061405429f, t, -1.453152027f);
    p = fmaf(p, t, 1.421413741f);
    p = fmaf(p, t, -0.284496736f);
    p = fmaf(p, t, 0.254829592f);
    p = p * t;
    // e^{-x^2} = 2^{-x^2 * log2(e)}  (v_exp_f32 = exp2)
    const float e = __builtin_amdgcn_exp2f(-(ax * ax) * 1.44269504089f);
    const float r = fmaf(-p, e, 1.0f);
    return copysignf(r, x);                                             // v_bfi
}

__global__ __launch_bounds__(KD_D) void kd_layernorm_kernel(
    const float* __restrict__ x,
    const float* __restrict__ weight,
    const float* __restrict__ bias,
    float* __restrict__ out)
{
    __shared__ float srow[KD_D];
    __shared__ float psum[8];
    __shared__ float psumsq[8];
    __shared__ float s_inv;

    const int row = blockIdx.x;
    const int tid = threadIdx.x;

    // ---- Stage row into LDS with CDNA5 async copy (ASYNCcnt-tracked) ------
    {
        const float* gp = x + (size_t)row * KD_D + tid;
        unsigned long long ga = (unsigned long long)(uintptr_t)gp;
        // Generic->LDS: low 32 bits of the flat address are the LDS offset.
        unsigned int lds = (unsigned int)(uintptr_t)(&srow[tid]);
        asm volatile("global_load_async_to_lds_b32 %0, %1, off"
                     :: "v"(lds), "v"(ga)
                     : "memory");
    }
#if __has_builtin(__builtin_amdgcn_s_wait_asynccnt)
    __builtin_amdgcn_s_wait_asynccnt(0);
#else
    asm volatile("s_wait_asynccnt 0" ::: "memory");
#endif
    __syncthreads();

    // ---- Row statistics: sum and sum of squares (wave32 shuffle tree) ----
    const float xi = srow[tid];
    float s1 = xi;
    float s2 = xi * xi;
#pragma unroll
    for (int off = 16; off > 0; off >>= 1) {
        s1 += __shfl_down(s1, off, 32);
        s2 += __shfl_down(s2, off, 32);
    }
    const int wid = tid >> 5;
    if ((tid & 31) == 0) { psum[wid] = s1; psumsq[wid] = s2; }
    __syncthreads();
    if (tid == 0) {
        float t1 = 0.0f, t2 = 0.0f;
#pragma unroll
        for (int w = 0; w < 8; ++w) { t1 += psum[w]; t2 += psumsq[w]; }
        // var with ddof=1
        const float var = (t2 - t1 * t1 * (1.0f / KD_D)) * (1.0f / (KD_D - 1));
        // bw = 0.9 * 256^-0.2 * std = 0.29688928 * std
        const float bw = 0.29688928f * sqrtf(var);
        s_inv = 0.70710678118f / bw;      // 1/(bw*sqrt(2))
    }
    __syncthreads();
    const float inv   = s_inv;
    const float ninv  = -inv;
    const float xiinv = xi * inv;         // (xi-xj)*inv == fma(-inv, xj, xi*inv)

    // ---- O(D^2) pairwise erf accumulation; LDS broadcast, float4 reads ----
    // 4 independent accumulators: no serial dependent-add chain, lets the
    // dual-issue VALU overlap adds with the next erf's FMA/TRANS chain.
    float a0 = 0.0f, a1 = 0.0f, a2 = 0.0f, a3 = 0.0f;
    const float4* s4 = reinterpret_cast<const float4*>(srow);
#pragma unroll 8
    for (int j = 0; j < KD_D / 4; ++j) {
        const float4 v = s4[j];
        a0 += fast_erf(fmaf(ninv, v.x, xiinv));
        a1 += fast_erf(fmaf(ninv, v.y, xiinv));
        a2 += fast_erf(fmaf(ninv, v.z, xiinv));
        a3 += fast_erf(fmaf(ninv, v.w, xiinv));
    }
    // 2*cdf - 1 = (1/D) * sum_j erf(...)
    const float u = ((a0 + a1) + (a2 + a3)) * (1.0f / KD_D);
    const float r = erfinvf(u) * 1.41421356237f;

    out[(size_t)row * KD_D + tid] = r * weight[tid] + bias[tid];
}

extern "C" void kernel_launch(void* const* d_in, const int* in_sizes, int n_in,
                              void* d_out, int out_size, void* d_ws, size_t ws_size,
                              hipStream_t stream) {
    const float* x      = (const float*)d_in[0];
    const float* weight = (const float*)d_in[1];
    const float* bias   = (const float*)d_in[2];
    float* out          = (float*)d_out;

    const int nrows = in_sizes[0] / KD_D;   // 4*1024 = 4096 rows
    kd_layernorm_kernel<<<nrows, KD_D, 0, stream>>>(x, weight, bias, out);
}